// EncoderDistillationLoss_19713899889321
// MI455X (gfx1250) — compile-verified
//
#include <hip/hip_runtime.h>

typedef __attribute__((ext_vector_type(2))) float v2f;
typedef __attribute__((ext_vector_type(8))) float v8f;

// Problem dims (fixed by the reference): B=8, C=512, T=2048, K=4096
// feats: (16384, 512) row i = (b,t), element c at sf[(b*512+c)*2048 + t]
// codebook: (4096, 512) row-major.  distance_matrix: (4096, 4096) row-major.

// symmetric online-softmax merge: (m,s,st,sl,bi) <- merge with (m2,s2,t2,l2,i2)
__device__ __forceinline__ void merge_stats(float& m, float& s, float& st, float& sl,
                                            int& bi, float m2, float s2, float t2,
                                            float l2, int i2) {
  const float mn = fmaxf(m, m2);
  const float e1 = __expf(m - mn), e2 = __expf(m2 - mn);
  s = s * e1 + s2 * e2;
  st = st * e1 + t2 * e2;
  sl = sl * e1 + l2 * e2;
  bi = (m >= m2) ? bi : i2;
  m = mn;
}

// ---------------------------------------------------------------------------
// Kernel 1: codebook row squared norms -> cnorm[4096]
// ---------------------------------------------------------------------------
__global__ __launch_bounds__(256) void cnorm_kernel(const float* __restrict__ cb,
                                                    float* __restrict__ cnorm) {
  const int warp = threadIdx.x >> 5, lane = threadIdx.x & 31;
  const int row = blockIdx.x * 8 + warp;                       // grid = 512
  const float4* p = (const float4*)(cb + (size_t)row * 512);
  float s = 0.f;
  for (int i = lane; i < 128; i += 32) {
    float4 v = p[i];
    s += v.x * v.x + v.y * v.y + v.z * v.z + v.w * v.w;
  }
#pragma unroll
  for (int off = 16; off; off >>= 1) s += __shfl_xor(s, off, 32);
  if (lane == 0) cnorm[row] = s;
}

// ---------------------------------------------------------------------------
// Kernel 2: feature MSE partial sums (per-block, deterministic)
// 8,388,608 floats = 2,097,152 float4 = 1024 blocks * 256 thr * 8
// ---------------------------------------------------------------------------
__global__ __launch_bounds__(256) void feat_mse_kernel(const float* __restrict__ sf,
                                                       const float* __restrict__ tf,
                                                       float* __restrict__ part) {
  __shared__ float sd[256];
  const float4* s4 = (const float4*)sf;
  const float4* t4 = (const float4*)tf;
  const int tid = threadIdx.x;
  size_t base = (size_t)blockIdx.x * 256 + tid;
  float acc = 0.f;
#pragma unroll
  for (int it = 0; it < 8; ++it) {
    size_t i = base + (size_t)it * 262144;
    float4 a = s4[i], b = t4[i];
    float dx = a.x - b.x, dy = a.y - b.y, dz = a.z - b.z, dw = a.w - b.w;
    acc += dx * dx + dy * dy + dz * dz + dw * dw;
  }
  sd[tid] = acc;
  __syncthreads();
  for (int s = 128; s > 0; s >>= 1) {
    if (tid < s) sd[tid] += sd[tid + s];
    __syncthreads();
  }
  if (tid == 0) part[blockIdx.x] = sd[0];
}

// ---------------------------------------------------------------------------
// Kernel 3: hard distance partials: dm[student[i], teacher[i]]  (64 blocks)
// ---------------------------------------------------------------------------
__global__ __launch_bounds__(256) void hard_kernel(const int* __restrict__ sc,
                                                   const int* __restrict__ tc,
                                                   const float* __restrict__ dm,
                                                   float* __restrict__ part) {
  __shared__ float sd[256];
  const int tid = threadIdx.x;
  const int i = blockIdx.x * 256 + tid;
  float v = dm[(size_t)sc[i] * 4096 + tc[i]];
  sd[tid] = v;
  __syncthreads();
  for (int s = 128; s > 0; s >>= 1) {
    if (tid < s) sd[tid] += sd[tid + s];
    __syncthreads();
  }
  if (tid == 0) part[blockIdx.x] = sd[0];
}

// ---------------------------------------------------------------------------
// Kernel 4: fused f32-WMMA GEMM + online softmax / argmin / entropy / E[td]
// grid = 1024 workgroups (16 rows each), 256 threads = 8 wave32.
// Per N-iteration the workgroup covers a 256-column stripe: each wave owns
// TWO 16x16 tiles (cols w*16 and w*16+128) sharing one A fragment, so each
// A ds-load feeds two v_wmma. 16 N-iterations cover 4096 codes.
// K = 512 staged in LDS chunks of 16.
// B LDS layout is [col][c] with pitch 20 floats: fragments are single aligned
// ds_load_b64 into even VGPR pairs, and the pitch tiles all 64 banks exactly
// once per half-wave (20*lm mod 64 = {0,4,...,60}) -> conflict-free.
// ---------------------------------------------------------------------------
#define BPITCH 20
__global__ __launch_bounds__(256) void main_kernel(const float* __restrict__ sf,
                                                   const int* __restrict__ tcodes,
                                                   const float* __restrict__ cb,
                                                   const float* __restrict__ dm,
                                                   const float* __restrict__ cnorm,
                                                   float* __restrict__ wsA) {
  __shared__ float Al[512 * 16];                    // A tile: [c][m], 32 KB
  __shared__ __align__(16) float Bl[256 * BPITCH];  // B chunk: [col][c], 20 KB
  __shared__ float fnorm[16];
  __shared__ int trow[16];
  __shared__ float red[256];
  __shared__ float wgM[128], wgS[128], wgT[128], wgL[128];
  __shared__ int wgI[128];

  const int tid = threadIdx.x;
  const int lane = tid & 31;
  const int wv = tid >> 5;
  const int lm = lane & 15;
  const int half = lane >> 4;

  const int rowbase = blockIdx.x * 16;
  const int batch = rowbase >> 11;          // 2048 rows per batch
  const int t0 = rowbase & 2047;
  const float* abase = sf + (size_t)batch * 512 * 2048 + t0;

  // ---- stage A (16 rows x 512 c), transposed to [c][m] ----
  for (int q = tid; q < 2048; q += 256) {
    const int c = q >> 2, m4 = (q & 3) << 2;
    const float4 v = *(const float4*)(abase + (size_t)c * 2048 + m4);
    Al[c * 16 + m4 + 0] = v.x;
    Al[c * 16 + m4 + 1] = v.y;
    Al[c * 16 + m4 + 2] = v.z;
    Al[c * 16 + m4 + 3] = v.w;
  }
  if (tid < 16) trow[tid] = tcodes[rowbase + tid];
  __syncthreads();

  // ---- per-row squared norms from LDS ----
  {
    const int row = tid & 15, seg = tid >> 4;
    float s = 0.f;
    for (int c = seg * 32; c < seg * 32 + 32; ++c) {
      float a = Al[c * 16 + row];
      s += a * a;
    }
    red[seg * 16 + row] = s;
  }
  __syncthreads();
  if (tid < 16) {
    float s = 0.f;
    for (int g = 0; g < 16; ++g) s += red[g * 16 + tid];
    fnorm[tid] = s;
  }
  __syncthreads();

  // ---- online-softmax state: 8 rows per lane (M = r + 8*half) ----
  float mM[8], sS[8], sT[8], sL[8];
  int bI[8];
#pragma unroll
  for (int r = 0; r < 8; ++r) {
    mM[r] = -__builtin_huge_valf();
    sS[r] = 0.f; sT[r] = 0.f; sL[r] = 0.f; bI[r] = 0;
  }

  const int colw0 = wv * 16 + lm;           // tile-0 column within stripe
  const int colw1 = colw0 + 128;            // tile-1 column within stripe

  for (int ni = 0; ni < 16; ++ni) {         // 16 * 256 = 4096 codes
    const int cwg = ni * 256;
    v8f acc0 = {};
    v8f acc1 = {};

    for (int kc = 0; kc < 512; kc += 16) {
      __syncthreads();                      // protect Bl reuse
      const bool pf = (kc + 16 < 512);
      // stage B chunk: 256 cols x 16 c, layout [col][c] (pitch 20)
      for (int q = tid; q < 1024; q += 256) {
        const int col = q >> 2, c4 = (q & 3) << 2;
        const float* src = cb + (size_t)(cwg + col) * 512 + kc + c4;
        const float4 v = *(const float4*)src;
        if (pf) __builtin_prefetch(src + 16, 0, 3);  // warm next chunk
        *(float4*)&Bl[col * BPITCH + c4] = v;
      }
      __syncthreads();

#pragma unroll
      for (int kk = 0; kk < 16; kk += 4) {
        const int kg = kc + kk;
        v2f a, b0, b1;
        // A 16x4 f32: lane m=lm, VGPR0/1 = K = 2*half + {0,1}   (ISA 7.12.2)
        a.x = Al[(kg + 2 * half + 0) * 16 + lm];
        a.y = Al[(kg + 2 * half + 1) * 16 + lm];
        // B 4x16 f32: lane n=lm, VGPR0/1 = K = 2*half + {0,1}; consecutive in LDS
        b0 = *(const v2f*)&Bl[colw0 * BPITCH + kk + 2 * half];
        b1 = *(const v2f*)&Bl[colw1 * BPITCH + kk + 2 * half];
        acc0 = __builtin_amdgcn_wmma_f32_16x16x4_f32(false, a, false, b0,
                                                     (short)0, acc0, false, false);
        acc1 = __builtin_amdgcn_wmma_f32_16x16x4_f32(false, a, false, b1,
                                                     (short)0, acc1, false, false);
      }
    }

    // ---- epilogue: both tiles update the same per-lane stats (fixed order) ----
#pragma unroll
    for (int g = 0; g < 2; ++g) {
      const int colj = cwg + (g ? colw1 : colw0);
      const float cn = cnorm[colj];
#pragma unroll
      for (int r = 0; r < 8; ++r) {
        const int row = r + 8 * half;       // C/D layout: VGPR r <-> M = r + 8*half
        const float dot = g ? acc1[r] : acc0[r];
        float sq = fnorm[row] + cn - 2.0f * dot;
        sq = fmaxf(sq, 0.0f);
        const float l = -sqrtf(sq);         // logit (TEMPERATURE == 1)
        const float td = dm[(size_t)trow[row] * 4096 + colj];
        if (l > mM[r]) {
          const float scl = __expf(mM[r] - l);
          sS[r] = sS[r] * scl + 1.0f;
          sT[r] = sT[r] * scl + td;
          sL[r] = sL[r] * scl + l;
          mM[r] = l;
          bI[r] = colj;
        } else {
          const float e = __expf(l - mM[r]);
          sS[r] += e;
          sT[r] += e * td;
          sL[r] += e * l;
        }
      }
    }
  }

  // ---- merge across the 16 lanes of each half-wave (columns of one row) ----
#pragma unroll
  for (int r = 0; r < 8; ++r) {
#pragma unroll
    for (int off = 8; off; off >>= 1) {     // masks 8,4,2,1 stay within 16-halves
      const float m2 = __shfl_xor(mM[r], off, 32);
      const float s2 = __shfl_xor(sS[r], off, 32);
      const float t2 = __shfl_xor(sT[r], off, 32);
      const float l2 = __shfl_xor(sL[r], off, 32);
      const int i2 = __shfl_xor(bI[r], off, 32);
      merge_stats(mM[r], sS[r], sT[r], sL[r], bI[r], m2, s2, t2, l2, i2);
    }
  }
  if (lm == 0) {                            // lanes 0 and 16
#pragma unroll
    for (int r = 0; r < 8; ++r) {
      const int row = r + 8 * half;
      wgM[wv * 16 + row] = mM[r];
      wgS[wv * 16 + row] = sS[r];
      wgT[wv * 16 + row] = sT[r];
      wgL[wv * 16 + row] = sL[r];
      wgI[wv * 16 + row] = bI[r];
    }
  }
  __syncthreads();

  // ---- merge across 8 waves (disjoint column stripes), fixed order ----
  if (tid < 16) {
    float m = wgM[tid], s = wgS[tid], st = wgT[tid], sl = wgL[tid];
    int bi = wgI[tid];
    for (int w = 1; w < 8; ++w) {
      merge_stats(m, s, st, sl, bi, wgM[w * 16 + tid], wgS[w * 16 + tid],
                  wgT[w * 16 + tid], wgL[w * 16 + tid], wgI[w * 16 + tid]);
    }
    const float inv = 1.0f / s;
    red[tid] = st * inv;                         // E[td] for this row
    red[16 + tid] = m + __logf(s) - sl * inv;    // entropy for this row
    red[32 + tid] = (bi == trow[tid]) ? 1.0f : 0.0f;
  }
  __syncthreads();
  if (tid == 0) {
    float a = 0.f, b = 0.f, c = 0.f;
    for (int i = 0; i < 16; ++i) { a += red[i]; b += red[16 + i]; c += red[32 + i]; }
    wsA[(size_t)blockIdx.x * 3 + 0] = a;
    wsA[(size_t)blockIdx.x * 3 + 1] = b;
    wsA[(size_t)blockIdx.x * 3 + 2] = c;
  }
}

// ---------------------------------------------------------------------------
// Kernel 5: deterministic final reduction + metric assembly
// ---------------------------------------------------------------------------
__global__ __launch_bounds__(256) void finalize_kernel(const float* __restrict__ wsA,
                                                       const float* __restrict__ wsB,
                                                       const float* __restrict__ wsC,
                                                       const float* __restrict__ vq,
                                                       float* __restrict__ out) {
  __shared__ float sd[256];
  const int tid = threadIdx.x;
  float f = 0.f, e = 0.f, n = 0.f, m = 0.f, h = 0.f;
  for (int i = tid; i < 1024; i += 256) {
    f += wsB[i];
    e += wsA[i * 3 + 0];
    n += wsA[i * 3 + 1];
    m += wsA[i * 3 + 2];
  }
  if (tid < 64) h = wsC[tid];
  float vals[5] = {f, e, n, m, h};
  float tot[5];
  for (int k = 0; k < 5; ++k) {
    sd[tid] = vals[k];
    __syncthreads();
    for (int s = 128; s > 0; s >>= 1) {
      if (tid < s) sd[tid] += sd[tid + s];
      __syncthreads();
    }
    tot[k] = sd[0];
    __syncthreads();
  }
  if (tid == 0) {
    const float feature = tot[0] / 8388608.0f;
    const float soft = tot[1] / 16384.0f;
    const float ent = tot[2] / 16384.0f;
    const float match = tot[3] / 16384.0f;
    const float hard = tot[4] / 16384.0f;
    const float vqv = vq[0];
    const float total = 1.0f * feature + 0.1f * soft + 0.0f * vqv;
    out[0] = total;   // tuple element 0
    out[1] = total;   // metrics[0]
    out[2] = feature;
    out[3] = soft;
    out[4] = hard;
    out[5] = vqv;
    out[6] = match;
    out[7] = ent;
  }
}

extern "C" void kernel_launch(void* const* d_in, const int* in_sizes, int n_in,
                              void* d_out, int out_size, void* d_ws, size_t ws_size,
                              hipStream_t stream) {
  (void)in_sizes; (void)n_in; (void)out_size; (void)ws_size;
  const float* sf = (const float*)d_in[0];
  const float* tf = (const float*)d_in[1];
  const int* tc = (const int*)d_in[2];
  const int* sc = (const int*)d_in[3];
  const float* vq = (const float*)d_in[4];
  const float* cb = (const float*)d_in[5];
  const float* dm = (const float*)d_in[6];
  float* out = (float*)d_out;
  float* ws = (float*)d_ws;

  float* cnorm = ws;            // 4096 floats
  float* wsB = ws + 4096;       // 1024 floats (feature-MSE partials)
  float* wsC = ws + 5120;       // 64 floats   (hard-distance partials)
  float* wsA = ws + 5184;       // 1024*3 floats (main-kernel partials)

  cnorm_kernel<<<512, 256, 0, stream>>>(cb, cnorm);
  feat_mse_kernel<<<1024, 256, 0, stream>>>(sf, tf, wsB);
  hard_kernel<<<64, 256, 0, stream>>>(sc, tc, dm, wsC);
  main_kernel<<<1024, 256, 0, stream>>>(sf, tc, cb, dm, cnorm, wsA);
  finalize_kernel<<<1, 256, 0, stream>>>(wsA, wsB, wsC, vq, out);
}